// KDS_67740224193293
// MI455X (gfx1250) — compile-verified
//
#include <hip/hip_runtime.h>
#include <math.h>

// KDS forward for MI455X (gfx1250, wave32).
// Phase 1 (kds_poca): per-polygon accelerated projected GD on 16-simplices.
//   - rank-2 restructure: x - step*grad = x + step*(y@A^T) - step*(x@(A A^T))
//   - per-layer x@(AA^T) done with V_WMMA_F32_16X16X4_F32 (4 chained WMMAs per
//     16-point tile), state held in WMMA C/D layout for all 20 layers.
//   - simplex projection = bitonic sort of 16 elems split across a lane pair
//     (L, L^16): strides 1/2/4 in-register, stride 8 via one shfl_xor per reg.
// Phase 2 (kds_fstep): per-point 65-simplex iteration; projection threshold
//   via Michelot fixed-point (exact Euclidean projection, no 65-wide sort).
// Workspace: points[G*N*2] + dist[G*N] floats = 50,331,648 bytes in d_ws.

#define KDS_N      65536
#define KDS_G      64
#define KDS_P      16
#define KDS_LAYERS 20

typedef float v2f __attribute__((ext_vector_type(2)));
typedef float v8f __attribute__((ext_vector_type(8)));

__device__ __forceinline__ float xor16(float x) { return __shfl_xor(x, 16, 32); }

__global__ void __launch_bounds__(256) kds_poca(
    const float* __restrict__ y, const float* __restrict__ W,
    const float* __restrict__ stepp, const int* __restrict__ polys,
    float* __restrict__ points, float* __restrict__ dist)
{
    __shared__ float Ash[KDS_P][2];
    const int g   = blockIdx.y;
    const int tid = threadIdx.x;
    if (tid < KDS_P) {
        const int idx = polys[g * KDS_P + tid];
        Ash[tid][0] = W[idx * 2 + 0];
        Ash[tid][1] = W[idx * 2 + 1];
    }
    __syncthreads();

    const float step = stepp[0];
    const int  lane = tid & 31;
    const int  pt   = lane & 15;          // point within 16-wide tile
    const bool hi   = (lane & 16) != 0;   // upper half-wave: comps 8..15
    const int  cb   = hi ? 8 : 0;         // component base for this lane
    const int  n    = (blockIdx.x * 8 + (tid >> 5)) * 16 + pt;

    const float y0 = y[2 * n + 0];
    const float y1 = y[2 * n + 1];

    // WMMA A operand: (-step)*(A A^T) in 16x4-chunk layout:
    // chunk j, VGPR v, lane L -> element (m = L%16, k = 4j + v + 2*(L>=16)).
    v2f a0, a1, a2, a3;
    {
        const float am0 = Ash[pt][0], am1 = Ash[pt][1];
        const int kb = hi ? 2 : 0;
        v2f aa[4];
#pragma unroll
        for (int j = 0; j < 4; ++j) {
#pragma unroll
            for (int v = 0; v < 2; ++v) {
                const int k = 4 * j + kb + v;
                aa[j][v] = -step * (am0 * Ash[k][0] + am1 * Ash[k][1]);
            }
        }
        a0 = aa[0]; a1 = aa[1]; a2 = aa[2]; a3 = aa[3];
    }

    // Loop-invariant bias sb = step*(y@A^T), already in C/D layout.
    float sb[8];
#pragma unroll
    for (int v = 0; v < 8; ++v)
        sb[v] = step * (y0 * Ash[cb + v][0] + y1 * Ash[cb + v][1]);

    // State in C/D layout: xt[v] = x_tmp[point pt][comp cb+v], xo = x_old.
    float xt[8], xo[8];
#pragma unroll
    for (int v = 0; v < 8; ++v) { xt[v] = 0.f; xo[v] = 0.f; }

    for (int layer = 0; layer < KDS_LAYERS; ++layer) {
        // Build B operand (X^T row-chunks k=4j..4j+3) from C/D-layout state:
        // B chunk j, VGPR v, lane L needs comp c = 4j + v + 2*(L>=16) of point
        // L%16, which lives in lane L or its half-wave partner.
        v2f b0, b1, b2, b3;
        {
            float s;
            s = xor16(xt[2]); b0[0] = hi ? s : xt[0];
            s = xor16(xt[3]); b0[1] = hi ? s : xt[1];
            s = xor16(xt[6]); b1[0] = hi ? s : xt[4];
            s = xor16(xt[7]); b1[1] = hi ? s : xt[5];
            s = xor16(xt[0]); b2[0] = hi ? xt[2] : s;
            s = xor16(xt[1]); b2[1] = hi ? xt[3] : s;
            s = xor16(xt[4]); b3[0] = hi ? xt[6] : s;
            s = xor16(xt[5]); b3[1] = hi ? xt[7] : s;
        }
        // z^T = (-step*AA^T) x X^T + (X_tmp + step*y@A^T): 4 chained WMMAs.
        v8f acc;
#pragma unroll
        for (int v = 0; v < 8; ++v) acc[v] = xt[v] + sb[v];
        acc = __builtin_amdgcn_wmma_f32_16x16x4_f32(false, a0, false, b0, (short)0, acc, false, false);
        acc = __builtin_amdgcn_wmma_f32_16x16x4_f32(false, a1, false, b1, (short)0, acc, false, false);
        acc = __builtin_amdgcn_wmma_f32_16x16x4_f32(false, a2, false, b2, (short)0, acc, false, false);
        acc = __builtin_amdgcn_wmma_f32_16x16x4_f32(false, a3, false, b3, (short)0, acc, false, false);

        float z[8], u[8];
#pragma unroll
        for (int v = 0; v < 8; ++v) { z[v] = acc[v]; u[v] = acc[v]; }

        // Bitonic descending sort of the 16 values held by lane pair (L,L^16);
        // global index e = v + 8*hi.
#define CEXD(i, jj) { float mx = fmaxf(u[i], u[jj]); float mn = fminf(u[i], u[jj]); u[i] = mx; u[jj] = mn; }
#define CEXA(i, jj) { float mx = fmaxf(u[i], u[jj]); float mn = fminf(u[i], u[jj]); u[i] = mn; u[jj] = mx; }
#define CEXR(i, jj) { float mx = fmaxf(u[i], u[jj]); float mn = fminf(u[i], u[jj]); u[i] = hi ? mn : mx; u[jj] = hi ? mx : mn; }
        CEXD(0,1) CEXA(2,3) CEXD(4,5) CEXA(6,7)              // k=2
        CEXD(0,2) CEXD(1,3) CEXA(4,6) CEXA(5,7)              // k=4 j=2
        CEXD(0,1) CEXD(2,3) CEXA(4,5) CEXA(6,7)              // k=4 j=1
        CEXR(0,4) CEXR(1,5) CEXR(2,6) CEXR(3,7)              // k=8 j=4 (dir=!hi)
        CEXR(0,2) CEXR(1,3) CEXR(4,6) CEXR(5,7)              // k=8 j=2
        CEXR(0,1) CEXR(2,3) CEXR(4,5) CEXR(6,7)              // k=8 j=1
#pragma unroll
        for (int v = 0; v < 8; ++v) {                        // k=16 j=8 (cross)
            const float p = xor16(u[v]);
            u[v] = hi ? fminf(u[v], p) : fmaxf(u[v], p);
        }
        CEXD(0,4) CEXD(1,5) CEXD(2,6) CEXD(3,7)              // k=16 j=4
        CEXD(0,2) CEXD(1,3) CEXD(4,6) CEXD(5,7)              // k=16 j=2
        CEXD(0,1) CEXD(2,3) CEXD(4,5) CEXD(6,7)              // k=16 j=1
#undef CEXD
#undef CEXA
#undef CEXR
        // cumsum over sorted u across the lane pair
        float cs[8];
        cs[0] = u[0];
#pragma unroll
        for (int v = 1; v < 8; ++v) cs[v] = cs[v - 1] + u[v];
        {
            const float low  = xor16(cs[7]);
            const float addv = hi ? low : 0.0f;
#pragma unroll
            for (int v = 0; v < 8; ++v) cs[v] += addv;
        }
        float vv[8];
#pragma unroll
        for (int v = 0; v < 8; ++v) vv[v] = (cs[v] - 1.0f) / (float)(v + cb + 1);
        float cnt = 0.f;
#pragma unroll
        for (int v = 0; v < 8; ++v) cnt += (u[v] > vv[v]) ? 1.f : 0.f;
        cnt += xor16(cnt);
        const int k  = (int)cnt - 1;
        const int kl = k - cb;
        float tl = vv[0];
#pragma unroll
        for (int v = 1; v < 8; ++v) tl = (kl == v) ? vv[v] : tl;
        const float tp  = xor16(tl);
        const float tau = (((k >> 3) != 0) == hi) ? tl : tp;

        const float beta = (float)layer / ((float)layer + 3.0f);
#pragma unroll
        for (int v = 0; v < 8; ++v) {
            const float fn = fmaxf(z[v] - tau, 0.0f);
            xt[v] = fn + beta * (fn - xo[v]);
            xo[v] = fn;
        }
    }

    // points = x_new @ A ; dist = |y - point| (Rprime = 1)
    float p0 = 0.f, p1 = 0.f;
#pragma unroll
    for (int v = 0; v < 8; ++v) {
        p0 += xo[v] * Ash[cb + v][0];
        p1 += xo[v] * Ash[cb + v][1];
    }
    p0 += xor16(p0);
    p1 += xor16(p1);
    if (!hi) {
        const size_t base = ((size_t)g * KDS_N + n) * 2;
        points[base + 0] = p0;
        points[base + 1] = p1;
        const float dx = y0 - p0, dy = y1 - p1;
        dist[(size_t)g * KDS_N + n] = sqrtf(dx * dx + dy * dy);
    }
}

__global__ void __launch_bounds__(128) kds_fstep(
    const float* __restrict__ y, const float* __restrict__ stepp,
    const float* __restrict__ points, const float* __restrict__ dist,
    float* __restrict__ out)
{
    __shared__ float gsh[65 * 128];   // step*grad per thread, bank-conflict-free
    const int tid = threadIdx.x;
    const int n   = blockIdx.x * 128 + tid;
    const float step = stepp[0];
    gsh[tid] = step;                  // grad[0] = 1
    for (int g = 0; g < KDS_G; ++g)
        gsh[(g + 1) * 128 + tid] = step * dist[(size_t)g * KDS_N + n];

    float w[65], fo[65];
#pragma unroll
    for (int i = 0; i < 65; ++i) { w[i] = 0.f; fo[i] = 0.f; }

    for (int layer = 0; layer < KDS_LAYERS; ++layer) {
#pragma unroll
        for (int i = 0; i < 65; ++i) w[i] -= gsh[i * 128 + tid];
        // Michelot fixed-point: tau rises monotonically to the exact simplex-
        // projection threshold (same result as the sort-based formula).
        float s = 0.f;
#pragma unroll
        for (int i = 0; i < 65; ++i) s += w[i];
        float tau = (s - 1.0f) * (1.0f / 65.0f);
        for (int it = 0; it < 64; ++it) {
            float s2 = 0.f, c2 = 0.f;
#pragma unroll
            for (int i = 0; i < 65; ++i) {
                const bool m = w[i] > tau;
                s2 += m ? w[i] : 0.f;
                c2 += m ? 1.f : 0.f;
            }
            if (c2 == 0.f) break;
            const float nt = (s2 - 1.0f) / c2;
            if (nt == tau) break;
            tau = nt;
        }
        const float beta = (float)layer / ((float)layer + 3.0f);
#pragma unroll
        for (int i = 0; i < 65; ++i) {
            const float fn = fmaxf(w[i] - tau, 0.0f);
            w[i] = fn + beta * (fn - fo[i]);
            fo[i] = fn;
        }
    }

    // Outputs: yhat [N,2] | f [N,64] | f0 [N]
    const float f0 = fo[0];
    float yh0 = y[2 * n + 0] * f0;
    float yh1 = y[2 * n + 1] * f0;
#pragma unroll
    for (int g = 0; g < KDS_G; ++g) {
        const float fg = fo[1 + g];
        const size_t base = ((size_t)g * KDS_N + n) * 2;
        yh0 += fg * points[base + 0];
        yh1 += fg * points[base + 1];
        out[(size_t)KDS_N * 2 + (size_t)n * KDS_G + g] = fg;
    }
    out[2 * n + 0] = yh0;
    out[2 * n + 1] = yh1;
    out[(size_t)KDS_N * 2 + (size_t)KDS_N * KDS_G + n] = f0;
}

extern "C" void kernel_launch(void* const* d_in, const int* in_sizes, int n_in,
                              void* d_out, int out_size, void* d_ws, size_t ws_size,
                              hipStream_t stream) {
    const float* y     = (const float*)d_in[0];
    const float* W     = (const float*)d_in[1];
    const float* step  = (const float*)d_in[2];
    const int*   polys = (const int*)d_in[3];
    float* out = (float*)d_out;

    float* points = (float*)d_ws;                        // G*N*2 floats
    float* dist   = points + (size_t)KDS_G * KDS_N * 2;  // G*N floats

    dim3 grid1(KDS_N / (16 * 8), KDS_G);                 // 8 waves/block, 16 pts/wave
    kds_poca<<<grid1, 256, 0, stream>>>(y, W, step, polys, points, dist);
    kds_fstep<<<KDS_N / 128, 128, 0, stream>>>(y, step, points, dist, out);
}